// FourierSparseAutoencoder_6966436954161
// MI455X (gfx1250) — compile-verified
//
#include <hip/hip_runtime.h>
#include <stdint.h>

typedef float v2f __attribute__((ext_vector_type(2)));
typedef float v8f __attribute__((ext_vector_type(8)));
typedef unsigned int u32x4 __attribute__((ext_vector_type(4)));
typedef int i32x4 __attribute__((ext_vector_type(4)));
typedef int i32x8 __attribute__((ext_vector_type(8)));

#define TWO_PI 6.28318530717958647692f

__device__ __forceinline__ v8f wmma4(v2f a, v2f b, v8f c) {
  // D = A(16x4,f32) * B(4x16,f32) + C(16x16,f32)
  return __builtin_amdgcn_wmma_f32_16x16x4_f32(false, a, false, b, (short)0, c,
                                               false, false);
}

__device__ __forceinline__ v8f v8zero() {
  v8f z = {0.f, 0.f, 0.f, 0.f, 0.f, 0.f, 0.f, 0.f};
  return z;
}

// DFT matrix elements from a 64-entry cosine table:
// C[i][j] = cos(2*pi*i*j/64); S[i][j] = sin(2*pi*i*j/64) = cos(2*pi*(i*j-16)/64)
__device__ __forceinline__ float Cel(const float* c64, int i, int j) {
  return c64[(i * j) & 63];
}
__device__ __forceinline__ float Sel(const float* c64, int i, int j) {
  return c64[((i * j) - 16) & 63];
}

// In-place: sX <- scale * (C @ sX @ C - S @ sX @ S)   [64x64, fp32 WMMA]
// This equals Re(FFT2(sX)) for scale=1 and Re(IFFT2(sX)) for scale=1/4096.
// Requires 256 threads (8 waves), sT/sU scratch 64x64 each.
__device__ __forceinline__ void dft2_real_inplace(float* sX, float* sT, float* sU,
                                                  const float* c64, float scale,
                                                  int t) {
  const int wid = t >> 5, lane = t & 31;
  const int lm = lane & 15;        // M or N index within tile
  const int lk = (lane >> 4) * 2;  // K sub-offset (0 or 2)
  const int hi = (lane >> 4);      // 0: rows 0-7, 1: rows 8-15 of C/D tile

  // ---- stage 1: T = C @ X ; U = S @ X  (32 output tiles over 8 waves) ----
#pragma unroll
  for (int i = 0; i < 4; ++i) {
    const int tile = wid * 4 + i;  // 0..31 (0-15 -> T, 16-31 -> U)
    const int tt = tile & 15;
    const int tm = (tt >> 2) * 16, tn = (tt & 3) * 16;
    const bool isT = (tile < 16);
    v8f acc = v8zero();
#pragma unroll
    for (int kk = 0; kk < 64; kk += 4) {
      const int k0 = kk + lk;
      v2f a, b;
      a.x = isT ? Cel(c64, tm + lm, k0) : Sel(c64, tm + lm, k0);
      a.y = isT ? Cel(c64, tm + lm, k0 + 1) : Sel(c64, tm + lm, k0 + 1);
      b.x = sX[k0 * 64 + tn + lm];
      b.y = sX[(k0 + 1) * 64 + tn + lm];
      acc = wmma4(a, b, acc);
    }
    float* dst = isT ? sT : sU;
#pragma unroll
    for (int r = 0; r < 8; ++r)
      dst[(tm + r + hi * 8) * 64 + tn + lm] = acc[r];
  }
  __syncthreads();

  // ---- stage 2: X = (T @ C - U @ S) * scale  (16 tiles over 8 waves) ----
#pragma unroll
  for (int i = 0; i < 2; ++i) {
    const int tile = wid * 2 + i;  // 0..15
    const int tm = (tile >> 2) * 16, tn = (tile & 3) * 16;
    v8f accP = v8zero(), accQ = v8zero();
#pragma unroll
    for (int kk = 0; kk < 64; kk += 4) {
      const int k0 = kk + lk;
      v2f a, b;
      a.x = sT[(tm + lm) * 64 + k0];
      a.y = sT[(tm + lm) * 64 + k0 + 1];
      b.x = Cel(c64, k0, tn + lm);
      b.y = Cel(c64, k0 + 1, tn + lm);
      accP = wmma4(a, b, accP);
      a.x = sU[(tm + lm) * 64 + k0];
      a.y = sU[(tm + lm) * 64 + k0 + 1];
      b.x = Sel(c64, k0, tn + lm);
      b.y = Sel(c64, k0 + 1, tn + lm);
      accQ = wmma4(a, b, accQ);
    }
#pragma unroll
    for (int r = 0; r < 8; ++r)
      sX[(tm + r + hi * 8) * 64 + tn + lm] = (accP[r] - accQ[r]) * scale;
  }
}

// ---------------- Kernel 1: f = Re(FFT2(x)) per image ----------------
__global__ __launch_bounds__(256) void fft_real_kernel(const float* __restrict__ x,
                                                       float* __restrict__ f) {
  __shared__ float sX[64 * 64];
  __shared__ float sT[64 * 64];
  __shared__ float sU[64 * 64];
  __shared__ float c64[64];
  const int t = threadIdx.x;
  const size_t b = blockIdx.x;
  if (t < 64) c64[t] = __cosf(TWO_PI * (float)t * (1.0f / 64.0f));
  const float* xb = x + b * 4096;
#pragma unroll
  for (int i = 0; i < 16; ++i) sX[t + 256 * i] = xb[t + 256 * i];
  __syncthreads();
  dft2_real_inplace(sX, sT, sU, c64, 1.0f, t);
  __syncthreads();
  float* fb = f + b * 4096;
#pragma unroll
  for (int i = 0; i < 16; ++i) fb[t + 256 * i] = sX[t + 256 * i];
}

// ---------------- TDM: 2D fp32 tile -> LDS (clang-23, 6-arg builtin) ---------
__device__ __forceinline__ void tdm_load_tile_f32(unsigned lds_byte_off,
                                                  const float* gptr,
                                                  unsigned tile_x, unsigned tile_y,
                                                  unsigned tensor_x, unsigned tensor_y,
                                                  unsigned row_stride) {
  const unsigned long long ga = (unsigned long long)(uintptr_t)gptr;
  u32x4 g0;
  g0[0] = 1u;                                                   // count=1, user D#
  g0[1] = lds_byte_off;                                         // lds_addr
  g0[2] = (unsigned)ga;                                         // global_addr[31:0]
  g0[3] = (unsigned)((ga >> 32) & 0x1FFFFFFull) | (2u << 30);   // [120:64] | type=2
  i32x8 g1;
  g1[0] = (int)(2u << 16);                                      // data_size=4B
  g1[1] = (int)((tensor_x & 0xFFFFu) << 16);                    // tensor_dim0 lo
  g1[2] = (int)(((tensor_x >> 16) & 0xFFFFu) |
                ((tensor_y & 0xFFFFu) << 16));                  // dim0 hi | dim1 lo
  g1[3] = (int)(((tensor_y >> 16) & 0xFFFFu) |
                ((tile_x & 0xFFFFu) << 16));                    // dim1 hi | tile_dim0
  g1[4] = (int)(tile_y & 0xFFFFu);                              // tile_dim1 (dim2=0)
  g1[5] = (int)row_stride;                                      // dim0_stride lo
  g1[6] = 0;                                                    // stride hi / dim1_stride
  g1[7] = 0;
  i32x4 gz4 = {0, 0, 0, 0};
  i32x8 gz8 = {0, 0, 0, 0, 0, 0, 0, 0};
  __builtin_amdgcn_tensor_load_to_lds(g0, g1, gz4, gz4, gz8, 0);
}

// ---------------- Kernel 2: z = f @ W1 + b1  (4096x2048x4096, fp32 WMMA) -----
// 128x128 block tile, K-chunk 32, A/B tiles staged by the Tensor Data Mover.
__global__ __launch_bounds__(256) void gemm1_kernel(const float* __restrict__ f,
                                                    const float* __restrict__ W1,
                                                    const float* __restrict__ b1,
                                                    float* __restrict__ z) {
  __shared__ float sA[128 * 32];  // f tile, row-major [m][k]
  __shared__ float sB[32 * 128];  // W1 tile, row-major [k][n]
  const int t = threadIdx.x, wid = t >> 5, lane = t & 31;
  const int lm = lane & 15, lk = (lane >> 4) * 2, hi = lane >> 4;
  const int n0 = blockIdx.x * 128, m0 = blockIdx.y * 128;
  const int wm = (wid & 3) * 32;   // wave M offset (4 waves in M)
  const int wn = (wid >> 2) * 64;  // wave N offset (2 waves in N)

  v8f acc[2][4];
#pragma unroll
  for (int mt = 0; mt < 2; ++mt)
#pragma unroll
    for (int nt = 0; nt < 4; ++nt) acc[mt][nt] = v8zero();

  const unsigned ldsA = (unsigned)(uintptr_t)(void*)sA;
  const unsigned ldsB = (unsigned)(uintptr_t)(void*)sB;

  for (int k0 = 0; k0 < 4096; k0 += 32) {
    __syncthreads();  // previous chunk's compute done before overwrite
    if (wid == 0) {
      // A: rows m0..m0+127, cols k0..k0+31 of f[4096x4096]
      tdm_load_tile_f32(ldsA, f + (size_t)m0 * 4096 + k0, 32, 128, 4096, 4096, 4096);
      // B: rows k0..k0+31, cols n0..n0+127 of W1[4096x2048]
      tdm_load_tile_f32(ldsB, W1 + (size_t)k0 * 2048 + n0, 128, 32, 2048, 4096, 2048);
      __builtin_amdgcn_s_wait_tensorcnt(0);
    }
    __syncthreads();  // LDS tiles visible to all waves
#pragma unroll
    for (int kk = 0; kk < 32; kk += 4) {
      const int kx = kk + lk;
      v2f a[2], bb[4];
#pragma unroll
      for (int mt = 0; mt < 2; ++mt) {
        const float* ap = &sA[(wm + mt * 16 + lm) * 32 + kx];
        a[mt].x = ap[0];
        a[mt].y = ap[1];
      }
#pragma unroll
      for (int nt = 0; nt < 4; ++nt) {
        const int col = wn + nt * 16 + lm;
        bb[nt].x = sB[kx * 128 + col];
        bb[nt].y = sB[(kx + 1) * 128 + col];
      }
#pragma unroll
      for (int mt = 0; mt < 2; ++mt)
#pragma unroll
        for (int nt = 0; nt < 4; ++nt)
          acc[mt][nt] = wmma4(a[mt], bb[nt], acc[mt][nt]);
    }
  }
  // epilogue: + b1, store
#pragma unroll
  for (int mt = 0; mt < 2; ++mt)
#pragma unroll
    for (int nt = 0; nt < 4; ++nt)
#pragma unroll
      for (int r = 0; r < 8; ++r) {
        const int row = m0 + wm + mt * 16 + r + hi * 8;
        const int col = n0 + wn + nt * 16 + lm;
        z[(size_t)row * 2048 + col] = acc[mt][nt][r] + b1[col];
      }
}

// ---------------- Kernel 3: per-row top-64 of |z|, compacted (val,idx) -------
__global__ __launch_bounds__(256) void topk_kernel(const float* __restrict__ z,
                                                   float* __restrict__ tv,
                                                   int* __restrict__ ti) {
  __shared__ unsigned keys[2048];
  __shared__ float vals[2048];
  __shared__ int red[256];
  __shared__ int pos;
  const int t = threadIdx.x;
  const size_t b = blockIdx.x;
  const float* zb = z + b * 2048;
#pragma unroll
  for (int i = 0; i < 8; ++i) {
    const float v = zb[t + 256 * i];
    vals[t + 256 * i] = v;
    keys[t + 256 * i] = __float_as_uint(v) & 0x7FFFFFFFu;  // |v| bits, monotonic
  }
  if (t == 0) pos = 0;
  __syncthreads();
  // bisect on the uint key for the 64th-largest value:
  // invariant: count(>=lo) >= 64, count(>=hi) < 64
  unsigned long long lo = 0ull, hi = 0x80000000ull;
  while (hi - lo > 1ull) {
    const unsigned mid = (unsigned)((lo + hi) >> 1);
    int c = 0;
#pragma unroll
    for (int i = 0; i < 8; ++i) c += (keys[t + 256 * i] >= mid) ? 1 : 0;
    red[t] = c;
    __syncthreads();
    for (int s = 128; s > 0; s >>= 1) {
      if (t < s) red[t] += red[t + s];
      __syncthreads();
    }
    const int total = red[0];
    __syncthreads();
    if (total >= 64) lo = mid; else hi = mid;
  }
  const unsigned kth = (unsigned)lo;
#pragma unroll
  for (int i = 0; i < 8; ++i) {
    const int e = t + 256 * i;
    if (keys[e] >= kth) {
      const int p = atomicAdd(&pos, 1);
      if (p < 64) {
        tv[b * 64 + p] = vals[e];
        ti[b * 64 + p] = e;
      }
    }
  }
}

// ---------------- Kernel 4: r = z_sparse @ W2 + b2 (sparse gather, 32x fewer FLOPs)
__global__ __launch_bounds__(256) void decoder_kernel(const float* __restrict__ tv,
                                                      const int* __restrict__ ti,
                                                      const float* __restrict__ W2,
                                                      const float* __restrict__ b2,
                                                      float* __restrict__ r) {
  __shared__ float sv[64];
  __shared__ int si[64];
  const int t = threadIdx.x;
  const size_t b = blockIdx.x;
  if (t < 64) {
    sv[t] = tv[b * 64 + t];
    si[t] = ti[b * 64 + t];
  }
  __syncthreads();
  float4 acc[4];
  const float4* b2v = (const float4*)b2;
#pragma unroll
  for (int j = 0; j < 4; ++j) acc[j] = b2v[t + 256 * j];
  for (int k = 0; k < 64; ++k) {
    const float v = sv[k];
    const float4* w = (const float4*)(W2 + (size_t)si[k] * 4096);  // L2-resident
#pragma unroll
    for (int j = 0; j < 4; ++j) {
      const float4 ww = w[t + 256 * j];
      acc[j].x += v * ww.x;
      acc[j].y += v * ww.y;
      acc[j].z += v * ww.z;
      acc[j].w += v * ww.w;
    }
  }
  float4* rv = (float4*)(r + b * 4096);
#pragma unroll
  for (int j = 0; j < 4; ++j) rv[t + 256 * j] = acc[j];
}

// ---------------- Kernel 5: Re(IFFT2(r)) then fused 3x3 SAME conv + bias -----
__global__ __launch_bounds__(256) void ifft_conv_kernel(const float* __restrict__ r,
                                                        const float* __restrict__ conv_w,
                                                        const float* __restrict__ conv_b,
                                                        float* __restrict__ out) {
  __shared__ float sX[64 * 64];
  __shared__ float sT[64 * 64];
  __shared__ float sU[64 * 64];
  __shared__ float c64[64];
  const int t = threadIdx.x;
  const size_t b = blockIdx.x;
  if (t < 64) c64[t] = __cosf(TWO_PI * (float)t * (1.0f / 64.0f));
  const float* rb = r + b * 4096;
#pragma unroll
  for (int i = 0; i < 16; ++i) sX[t + 256 * i] = rb[t + 256 * i];
  __syncthreads();
  dft2_real_inplace(sX, sT, sU, c64, 1.0f / 4096.0f, t);
  __syncthreads();
  float w[9];
#pragma unroll
  for (int i = 0; i < 9; ++i) w[i] = conv_w[i];
  const float cb = conv_b[0];
  float* ob = out + b * 4096;
#pragma unroll
  for (int i = 0; i < 16; ++i) {
    const int p = t + 256 * i, pi = p >> 6, pj = p & 63;
    float s = cb;
#pragma unroll
    for (int kh = 0; kh < 3; ++kh) {
      const int ii = pi + kh - 1;
      if (ii < 0 || ii > 63) continue;
#pragma unroll
      for (int kw = 0; kw < 3; ++kw) {
        const int jj = pj + kw - 1;
        if (jj < 0 || jj > 63) continue;
        s += w[kh * 3 + kw] * sX[ii * 64 + jj];
      }
    }
    ob[p] = s;
  }
}

extern "C" void kernel_launch(void* const* d_in, const int* in_sizes, int n_in,
                              void* d_out, int out_size, void* d_ws, size_t ws_size,
                              hipStream_t stream) {
  (void)in_sizes; (void)n_in; (void)out_size; (void)ws_size;
  const float* x  = (const float*)d_in[0];   // [4096,1,64,64]
  const float* W1 = (const float*)d_in[1];   // [4096,2048]
  const float* b1 = (const float*)d_in[2];   // [2048]
  const float* W2 = (const float*)d_in[3];   // [2048,4096]
  const float* b2 = (const float*)d_in[4];   // [4096]
  const float* cw = (const float*)d_in[5];   // [1,1,3,3]
  const float* cb = (const float*)d_in[6];   // [1]
  float* out = (float*)d_out;                // [4096,1,64,64]

  // workspace layout (floats): f/r (reused) | z | topk vals | topk idx  ~98MB
  float* f_buf = (float*)d_ws;                       // 4096*4096
  float* z_buf = f_buf + (size_t)4096 * 4096;        // 4096*2048
  float* tv    = z_buf + (size_t)4096 * 2048;        // 4096*64
  int*   ti    = (int*)(tv + (size_t)4096 * 64);     // 4096*64

  fft_real_kernel<<<4096, 256, 0, stream>>>(x, f_buf);
  gemm1_kernel<<<dim3(2048 / 128, 4096 / 128), 256, 0, stream>>>(f_buf, W1, b1, z_buf);
  topk_kernel<<<4096, 256, 0, stream>>>(z_buf, tv, ti);
  decoder_kernel<<<4096, 256, 0, stream>>>(tv, ti, W2, b2, f_buf);  // r reuses f_buf
  ifft_conv_kernel<<<4096, 256, 0, stream>>>(f_buf, cw, cb, out);
}